// TopKGate_87651692577399
// MI455X (gfx1250) — compile-verified
//
#include <hip/hip_runtime.h>
#include <hip/hip_bf16.h>

typedef __attribute__((ext_vector_type(2))) float v2f;
typedef __attribute__((ext_vector_type(4))) float v4f;
typedef __attribute__((ext_vector_type(8))) float v8f;

#define D_MODEL 4096
#define N_EXPERT 64
#define N_TOKENS 16384

__device__ __forceinline__ v8f wmma_f32(v2f a, v2f b, v8f c) {
    // 8 args: (neg_a, A, neg_b, B, c_mod, C, reuse_a, reuse_b)
    return __builtin_amdgcn_wmma_f32_16x16x4_f32(false, a, false, b, (short)0, c,
                                                 false, false);
}

// One wave computes a 32-token x 64-expert tile of logits:
//   - x (256 MiB stream) is read from HBM exactly once, with NT hint so the
//     stream does not evict the hot W working set from L2.
//   - W (1 MiB) stays resident in L2/WGP$ (regular temporal loads).
//   - Per 8-K step: 2 A-loads + 4 B-loads (b128) feed 16 WMMAs; dependent
//     D->C WMMA pairs are separated by 7 independent WMMAs.
__global__ __launch_bounds__(64) void moe_gemm_kernel(
        const float* __restrict__ x, const float* __restrict__ W,
        float* __restrict__ logits) {
    const int lane = threadIdx.x & 31;
    const int wid  = blockIdx.x * 2 + (threadIdx.x >> 5);  // 32-token tile id, 0..511
    const int tokBase = wid * 32;
    const int l15 = lane & 15;
    const int hi  = lane >> 4;   // 0: K-slots {0,1}; 1: K-slots {2,3}

    // lo half-wave covers k..k+3, hi half-wave covers k+4..k+7 each 8-K step.
    const float* aPtr0 = x + (size_t)(tokBase +  0 + l15) * D_MODEL + hi * 4;
    const float* aPtr1 = x + (size_t)(tokBase + 16 + l15) * D_MODEL + hi * 4;
    const float* bPtr  = W + (size_t)l15 * D_MODEL + hi * 4;   // expert tiles via +16*D_MODEL

    v8f acc[2][4];
#pragma unroll
    for (int t = 0; t < 2; ++t)
#pragma unroll
        for (int e = 0; e < 4; ++e) acc[t][e] = (v8f){};

    for (int k = 0; k < D_MODEL; k += 8) {
        // x: non-temporal stream
        v4f av0 = __builtin_nontemporal_load((const v4f*)(aPtr0 + k));
        v4f av1 = __builtin_nontemporal_load((const v4f*)(aPtr1 + k));
        // W: temporal (cache-resident)
        v4f bv0 = *(const v4f*)(bPtr + k + (size_t)0 * 16 * D_MODEL);
        v4f bv1 = *(const v4f*)(bPtr + k + (size_t)1 * 16 * D_MODEL);
        v4f bv2 = *(const v4f*)(bPtr + k + (size_t)2 * 16 * D_MODEL);
        v4f bv3 = *(const v4f*)(bPtr + k + (size_t)3 * 16 * D_MODEL);

        v2f a00 = av0.xy, a01 = av0.zw;   // token sub-tile 0
        v2f a10 = av1.xy, a11 = av1.zw;   // token sub-tile 1

        // K-slots (k,k+1 | k+4,k+5)
        acc[0][0] = wmma_f32(a00, bv0.xy, acc[0][0]);
        acc[1][0] = wmma_f32(a10, bv0.xy, acc[1][0]);
        acc[0][1] = wmma_f32(a00, bv1.xy, acc[0][1]);
        acc[1][1] = wmma_f32(a10, bv1.xy, acc[1][1]);
        acc[0][2] = wmma_f32(a00, bv2.xy, acc[0][2]);
        acc[1][2] = wmma_f32(a10, bv2.xy, acc[1][2]);
        acc[0][3] = wmma_f32(a00, bv3.xy, acc[0][3]);
        acc[1][3] = wmma_f32(a10, bv3.xy, acc[1][3]);
        // K-slots (k+2,k+3 | k+6,k+7)
        acc[0][0] = wmma_f32(a01, bv0.zw, acc[0][0]);
        acc[1][0] = wmma_f32(a11, bv0.zw, acc[1][0]);
        acc[0][1] = wmma_f32(a01, bv1.zw, acc[0][1]);
        acc[1][1] = wmma_f32(a11, bv1.zw, acc[1][1]);
        acc[0][2] = wmma_f32(a01, bv2.zw, acc[0][2]);
        acc[1][2] = wmma_f32(a11, bv2.zw, acc[1][2]);
        acc[0][3] = wmma_f32(a01, bv3.zw, acc[0][3]);
        acc[1][3] = wmma_f32(a11, bv3.zw, acc[1][3]);
    }

    // C/D layout: VGPR j holds M=j (lanes 0-15) / M=j+8 (lanes 16-31), N = lane&15
#pragma unroll
    for (int t = 0; t < 2; ++t) {
        float* outBase = logits + (size_t)(tokBase + t * 16) * N_EXPERT;
#pragma unroll
        for (int j = 0; j < 8; ++j) {
            const int m = j + hi * 8;
            float* row = outBase + m * N_EXPERT + l15;
            row[0]  = acc[t][0][j];
            row[16] = acc[t][1][j];
            row[32] = acc[t][2][j];
            row[48] = acc[t][3][j];
        }
    }
}

// One thread per token: top-2 (jax tie-break: lower index wins on equal),
// 2-way softmax scatter, logsumexp^2.
__global__ __launch_bounds__(256) void gate_epilogue_kernel(
        const float* __restrict__ logits, float* __restrict__ scores,
        float* __restrict__ loss) {
    const int t = blockIdx.x * blockDim.x + threadIdx.x;
    const float* row = logits + (size_t)t * N_EXPERT;

    float r[N_EXPERT];
#pragma unroll
    for (int i = 0; i < N_EXPERT / 4; ++i) {
        v4f v = *(const v4f*)(row + 4 * i);
        r[4 * i + 0] = v.x;
        r[4 * i + 1] = v.y;
        r[4 * i + 2] = v.z;
        r[4 * i + 3] = v.w;
    }

    float m1 = -3.402823466e+38f, m2 = -3.402823466e+38f;
    int i1 = 0, i2 = 0;
#pragma unroll
    for (int e = 0; e < N_EXPERT; ++e) {
        float v = r[e];
        if (v > m1) { m2 = m1; i2 = i1; m1 = v; i1 = e; }
        else if (v > m2) { m2 = v; i2 = e; }
    }

    float sum = 0.0f;
#pragma unroll
    for (int e = 0; e < N_EXPERT; ++e) sum += __expf(r[e] - m1);
    const float log_z = m1 + __logf(sum);
    loss[t] = log_z * log_z;

    // softmax over {m1, m2}, m1 >= m2
    const float e2  = __expf(m2 - m1);
    const float inv = 1.0f / (1.0f + e2);
    const float s1 = inv, s2 = e2 * inv;

    float* srow = scores + (size_t)t * N_EXPERT;
#pragma unroll
    for (int i = 0; i < N_EXPERT / 4; ++i) {
        v4f o;
        const int e = 4 * i;
        o.x = (e + 0 == i1) ? s1 : ((e + 0 == i2) ? s2 : 0.0f);
        o.y = (e + 1 == i1) ? s1 : ((e + 1 == i2) ? s2 : 0.0f);
        o.z = (e + 2 == i1) ? s1 : ((e + 2 == i2) ? s2 : 0.0f);
        o.w = (e + 3 == i1) ? s1 : ((e + 3 == i2) ? s2 : 0.0f);
        *(v4f*)(srow + e) = o;
    }
}

extern "C" void kernel_launch(void* const* d_in, const int* in_sizes, int n_in,
                              void* d_out, int out_size, void* d_ws, size_t ws_size,
                              hipStream_t stream) {
    const float* x = (const float*)d_in[0];   // (4,4096,4096) f32
    const float* W = (const float*)d_in[1];   // (64,4096) f32

    float* out    = (float*)d_out;
    float* logits = out;                                // 16384*64
    float* scores = out + (size_t)N_TOKENS * N_EXPERT;  // 16384*64
    float* loss   = out + 2 * (size_t)N_TOKENS * N_EXPERT;

    // 512 tiles of 32 tokens; 2 waves (64 threads) per block -> 256 blocks
    moe_gemm_kernel<<<256, 64, 0, stream>>>(x, W, logits);
    // 16384 tokens, 256 threads/block
    gate_epilogue_kernel<<<64, 256, 0, stream>>>(logits, scores, loss);
}